// RoIPooling2D_74715251081426
// MI455X (gfx1250) — compile-verified
//
#include <hip/hip_runtime.h>

// ---------------------------------------------------------------------------
// RoI max pooling for MI455X (gfx1250) using the Tensor Data Mover (TDM).
//
// One workgroup (256 threads = 8 wave32) per ROI.  The ROI's feature-map
// region (w_roi x h_roi, up to 64x64) is DMA'd channel-chunk by channel-chunk
// from global memory into LDS with tensor_load_to_lds (3D tile: X=w, Y=h,
// Z=channel), double-buffered and tracked with TENSORcnt.  Compute waves do
// the 7x7 bin max-reductions out of LDS while the next chunk streams in.
// ---------------------------------------------------------------------------

typedef unsigned int v4u __attribute__((ext_vector_type(4)));
typedef int          v8i __attribute__((ext_vector_type(8)));
typedef int          v4i __attribute__((ext_vector_type(4)));

namespace {
constexpr int kC = 256;
constexpr int kH = 64;
constexpr int kW = 64;
constexpr int kOut = 7;
constexpr int kBufFloats = 8192;            // 32 KB per buffer (2 buffers = 64 KB LDS)
constexpr int kBufBytes  = kBufFloats * 4;
}

__device__ __forceinline__ int rfl(int x) { return __builtin_amdgcn_readfirstlane(x); }

// Build a D# for a 3-D tile (tile_dim0=w_roi, tile_dim1=h_roi, tile_dim2=cc)
// of a tensor with dim0=W, dim1=H, strides {W, H*W, H*W} (f32 elements), and
// fire TENSOR_LOAD_TO_LDS.  All descriptor dwords are wave-uniform.
__device__ __forceinline__ void tdm_load_tile(unsigned long long gaddr,
                                              unsigned lds_byte_off,
                                              int cc, int h_roi, int w_roi)
{
  v4u g0;
  g0.x = 1u;                                                    // count = 1, user mode
  g0.y = (unsigned)rfl((int)lds_byte_off);                      // lds_addr (bytes)
  g0.z = (unsigned)rfl((int)(unsigned)(gaddr & 0xffffffffull)); // global_addr[31:0]
  g0.w = ((unsigned)rfl((int)(unsigned)(gaddr >> 32)) & 0x01ffffffu)
         | 0x80000000u;                                         // addr[56:32] | type=2

  v8i g1;
  g1[0] = 2 << 16;                          // data_size=2 (4B); no multicast/pad/iterate
  g1[1] = (kW & 0xffff) << 16;              // tensor_dim0[15:0]  = W
  g1[2] = (kH & 0xffff) << 16;              // dim0[31:16]=0 | tensor_dim1[15:0] = H
  g1[3] = rfl((w_roi & 0xffff) << 16);      // dim1[31:16]=0 | tile_dim0 = w_roi
  g1[4] = rfl((h_roi & 0xffff) | ((cc & 0xffff) << 16)); // tile_dim1 | tile_dim2
  g1[5] = kW;                               // tensor_dim0_stride[31:0] = W
  g1[6] = ((kH * kW) & 0xffff) << 16;       // stride0[47:32]=0 | stride1[15:0]
  g1[7] = (kH * kW) >> 16;                  // tensor_dim1_stride[47:16]

  v4i g2;
  g2[0] = rfl(cc);                          // tensor_dim2 (== tile depth; never OOB)
  g2[1] = 0;                                // tensor_dim3 (unused)
  g2[2] = kH * kW;                          // tensor_dim2_stride[31:0] = plane stride
  g2[3] = 0;                                // stride2[47:32]=0 | tile_dim3=0

  v4i g3 = {0, 0, 0, 0};

  // clang-23 / therock signature: (v4u, v8i, v4i, v4i, v8i, i32 cpol).
  // The trailing 8-dword group is unused for a normal (non-gather,
  // non-iterate) <=4D descriptor; pass zeros.
  v8i g4 = {0, 0, 0, 0, 0, 0, 0, 0};

  __builtin_amdgcn_tensor_load_to_lds(g0, g1, g2, g3, g4, /*cpol=*/0);
}

__global__ __launch_bounds__(256)
void roi_pool_tdm(const float* __restrict__ fm,
                  const int*   __restrict__ rois,
                  float*       __restrict__ out)
{
  __shared__ float tile[2 * kBufFloats];
  // Low 32 bits of the generic pointer to an LDS object are its byte offset
  // within the workgroup's LDS allocation.  The ptr->int cast also escapes
  // `tile` so the compiler cannot assume it is never written (TDM writes it).
  const unsigned lds_base = (unsigned)(unsigned long long)(void*)&tile[0];

  const int roi = blockIdx.x;
  const int b   = rois[roi * 5 + 0];
  const int ltx = rois[roi * 5 + 1] >> 4;   // *1/16 is exact in f32; trunc == >>4
  const int lty = rois[roi * 5 + 2] >> 4;
  const int rbx = rois[roi * 5 + 3] >> 4;
  const int rby = rois[roi * 5 + 4] >> 4;
  const int w_roi = rbx - ltx + 1;          // 1..64
  const int h_roi = rby - lty + 1;          // 1..64
  const int area  = w_roi * h_roi;          // <= 4096 floats

  int cc = kBufFloats / area;               // channels per chunk, >= 2 worst case
  if (cc > kC) cc = kC;
  const int nchunks = (kC + cc - 1) / cc;

  const unsigned long long base_addr =
      (unsigned long long)fm +
      4ull * ((((unsigned long long)(unsigned)b * kC) * kH + (unsigned)lty) * kW
              + (unsigned)ltx);
  const unsigned long long chan_bytes = 4ull * kH * kW;

  const bool leader = (rfl((int)(threadIdx.x >> 5)) == 0);  // wave 0 only

  if (leader) {
    const int cc0 = (kC < cc) ? kC : cc;
    tdm_load_tile(base_addr, lds_base, cc0, h_roi, w_roi);
  }

  for (int k = 0; k < nchunks; ++k) {
    const int c0  = k * cc;
    const int cck = (kC - c0 < cc) ? (kC - c0) : cc;
    const int buf = k & 1;

    if (leader) __builtin_amdgcn_s_wait_tensorcnt(0);  // chunk k resident in LDS
    __syncthreads();                                   // publish to all waves
    asm volatile("" ::: "memory");                     // force LDS reloads below

    // Prefetch chunk k+1 into the other buffer while we compute chunk k.
    // Safe: every thread passed the barrier above, so nobody still reads buf^1.
    if (leader && (k + 1) < nchunks) {
      const int c1  = (k + 1) * cc;
      const int cc1 = (kC - c1 < cc) ? (kC - c1) : cc;
      tdm_load_tile(base_addr + chan_bytes * (unsigned long long)c1,
                    lds_base + (unsigned)(buf ^ 1) * kBufBytes,
                    cc1, h_roi, w_roi);
    }

    const float* bufp  = &tile[buf * kBufFloats];
    const int    items = cck * (kOut * kOut);
    for (int it = (int)threadIdx.x; it < items; it += 256) {
      const int cl  = it / 49;
      const int bin = it - cl * 49;
      const int i = bin / 7;
      const int j = bin - i * 7;
      // Bin bounds relative to (lty, ltx); bins are always non-empty.
      const int hs = (i * h_roi) / 7;
      const int he = ((i + 1) * h_roi + 6) / 7;
      const int ws = (j * w_roi) / 7;
      const int we = ((j + 1) * w_roi + 6) / 7;

      const float* plane = bufp + cl * area;
      float m = -__builtin_inff();
      for (int h = hs; h < he; ++h) {
        const float* row = plane + h * w_roi;
        for (int w = ws; w < we; ++w) m = fmaxf(m, row[w]);
      }
      out[(((unsigned long long)roi * kC + (unsigned)(c0 + cl)) * kOut + i) * kOut + j] = m;
    }
    // No trailing barrier needed: next iteration's leading barrier orders
    // everything before the next TDM issue into this buffer.
  }
}

extern "C" void kernel_launch(void* const* d_in, const int* in_sizes, int n_in,
                              void* d_out, int out_size, void* d_ws, size_t ws_size,
                              hipStream_t stream) {
  (void)n_in; (void)out_size; (void)d_ws; (void)ws_size;
  const float* fm   = (const float*)d_in[0];
  const int*   rois = (const int*)d_in[1];
  float*       out  = (float*)d_out;
  const int N = in_sizes[1] / 5;   // rois is (N,5)
  roi_pool_tdm<<<dim3(N), dim3(256), 0, stream>>>(fm, rois, out);
}